// LinearModel_28604482191491
// MI455X (gfx1250) — compile-verified
//
#include <hip/hip_runtime.h>
#include <math.h>

// DLRM-style fused kernel for gfx1250 (MI455X, wave32):
//   out[b] = sigmoid( sum_{c,l} cat_tables[c][cat_ids[c,b,l]] + dense_x[b,:]·dense_w + bias )
//
// - Dense matvec exact in f32 via V_WMMA_F32_16X16X4_F32 (dense_w chunk
//   broadcast across all 16 N columns; lane N=0 extracts the matvec result).
// - Gathers: column loop is WAVE-UNIFORM so each table base is an SGPR and
//   gathers lower to the SADDR + 32-bit-voffset form (1 VALU/gather instead
//   of a 64-bit address build). Lane halves split the bag (l=0..9 vs 10..19)
//   of the SAME row, merged with one shfl_xor(16).
// - 32MB of tables stay L2-resident (192MB L2): regular-temporal gathers.
//   Streamed-once data (cat_ids, dense_x, out) uses non-temporal hints.

typedef __attribute__((ext_vector_type(2))) float    v2f;
typedef __attribute__((ext_vector_type(8))) float    v8f;
typedef __attribute__((ext_vector_type(2))) unsigned uv2;

#define CCOLS 8
#define BAG   20
#define HBAG  10
#define DDIM  256
#define WAVES_PER_BLOCK 2

__global__ __launch_bounds__(32 * WAVES_PER_BLOCK)
void dlrm_fused_kernel(const unsigned* __restrict__ cat_ids,     // [C,B,L] int32 (all >= 0)
                       const float*    __restrict__ cat_tables,  // [C,V]
                       const float*    __restrict__ dense_x,     // [B,D]
                       const float*    __restrict__ dense_w,     // [D]
                       const float*    __restrict__ bias,        // [1]
                       float*          __restrict__ out,         // [B]
                       int B, int V)
{
    __shared__ float dlog[WAVES_PER_BLOCK * 16];

    const int lane = threadIdx.x & 31;
    const int wid  = threadIdx.x >> 5;
    const int tile = blockIdx.x * WAVES_PER_BLOCK + wid;   // 16-row batch tile
    const int row0 = tile * 16;

    const int m    = lane & 15;           // batch row within tile (both halves)
    const int hi   = lane >> 4;           // 0: lanes 0-15, 1: lanes 16-31
    const int koff = hi << 1;             // WMMA A K-interleave: 0 or 2
    const int row  = row0 + m;
    const unsigned l0 = (unsigned)hi * HBAG;   // bag half: 0..9 or 10..19

    // Warm the index stream for this lane's bag halves (global_prefetch_b8).
    #pragma unroll
    for (int c = 0; c < CCOLS; ++c)
        __builtin_prefetch(cat_ids + ((unsigned)c * (unsigned)B + (unsigned)row) * BAG + l0, 0, 1);

    // ---- Dense matvec: 64 x V_WMMA_F32_16X16X4_F32, K = 0..255 ----
    // A (16x4 f32): lane m holds A[m][k+koff], A[m][k+koff+1]  (one float2 load)
    // B (4x16 f32): dense_w[k..k+3] broadcast across N, mirrored K interleave.
    const float* __restrict__ xrow = dense_x + (size_t)row * DDIM;
    v8f acc = {};
    #pragma unroll 8
    for (int k = 0; k < DDIM; k += 4) {
        v2f a = __builtin_nontemporal_load((const v2f*)(xrow + k + koff));
        v2f b = *(const v2f*)(dense_w + k + koff);   // tiny, reused by all waves: RT
        acc = __builtin_amdgcn_wmma_f32_16x16x4_f32(
                  /*neg_a=*/false, a, /*neg_b=*/false, b,
                  /*c_mod=*/(short)0, acc, /*reuse_a=*/false, /*reuse_b=*/false);
    }

    // ---- Embedding-bag gathers: uniform column loop -> SGPR table base ----
    // Per column: 5x b64 NT index loads (contiguous 40B run, 8B aligned),
    // then 10 independent SADDR-form gathers; two accumulators break the
    // serial FP chain so the scheduler can keep gathers in flight.
    float catsum = 0.0f;
    #pragma unroll
    for (int c = 0; c < CCOLS; ++c) {
        const float* __restrict__ tab = cat_tables + (size_t)c * (size_t)V;   // uniform => SGPR
        const unsigned ofs = ((unsigned)c * (unsigned)B + (unsigned)row) * BAG + l0;
        const uv2* __restrict__ idp = (const uv2*)(cat_ids + ofs);            // 8B aligned

        uv2 id2[HBAG / 2];
        #pragma unroll
        for (int i = 0; i < HBAG / 2; ++i)
            id2[i] = __builtin_nontemporal_load(idp + i);                     // streamed once

        float v[HBAG];
        #pragma unroll
        for (int i = 0; i < HBAG / 2; ++i) {
            v[2 * i + 0] = tab[(size_t)id2[i].x];   // zero-extended 32-bit voffset
            v[2 * i + 1] = tab[(size_t)id2[i].y];
        }

        float s0 = 0.0f, s1 = 0.0f;
        #pragma unroll
        for (int i = 0; i < HBAG; i += 2) { s0 += v[i]; s1 += v[i + 1]; }
        catsum += s0 + s1;
    }
    // Merge the two bag halves: lane m (<16) ends with the full row sum.
    catsum += __shfl_xor(catsum, 16, 32);

    // ---- Route WMMA D-matrix column N=0 to the row-owning lanes via LDS ----
    // Lane 0 holds D[M=0..7][0] in acc[0..7]; lane 16 holds D[M=8..15][0].
    if ((lane & 15) == 0) {
        const int base = wid * 16 + (hi << 3);
        #pragma unroll
        for (int i = 0; i < 8; ++i) dlog[base + i] = acc[i];
    }
    __syncthreads();

    if (lane < 16) {
        const float logit = dlog[wid * 16 + lane] + catsum + bias[0];
        const float sig   = 1.0f / (1.0f + __expf(-logit));
        __builtin_nontemporal_store(sig, out + row0 + lane);   // streamed once
    }
}

extern "C" void kernel_launch(void* const* d_in, const int* in_sizes, int n_in,
                              void* d_out, int out_size, void* d_ws, size_t ws_size,
                              hipStream_t stream) {
    const unsigned* cat_ids    = (const unsigned*)d_in[0];
    const float*    cat_tables = (const float*)   d_in[1];
    const float*    dense_x    = (const float*)   d_in[2];
    const float*    dense_w    = (const float*)   d_in[3];
    const float*    bias       = (const float*)   d_in[4];
    float*          out        = (float*)d_out;

    const int B = in_sizes[2] / DDIM;     // 16384
    const int V = in_sizes[1] / CCOLS;    // 1,000,000

    const int tiles  = B / 16;                         // 1024 (B divisible by 16)
    const int blocks = tiles / WAVES_PER_BLOCK;        // 512
    dlrm_fused_kernel<<<dim3(blocks), dim3(32 * WAVES_PER_BLOCK), 0, stream>>>(
        cat_ids, cat_tables, dense_x, dense_w, bias, out, B, V);
}